// GCN_90366111908651
// MI455X (gfx1250) — compile-verified
//
#include <hip/hip_runtime.h>
#include <math.h>

#define N_NODES 100000
#define N_EDGES 1600000
#define IN_FEAT 128
#define HID 64
#define CLS 40
#define CLS_PAD 48

typedef __attribute__((ext_vector_type(2))) float v2f;
typedef __attribute__((ext_vector_type(8))) float v8f;

__device__ __forceinline__ v8f wmma_f32_k4(v2f a, v2f b, v8f c) {
    // V_WMMA_F32_16X16X4_F32 : D = A(16x4) x B(4x16) + C(16x16), all f32
    return __builtin_amdgcn_wmma_f32_16x16x4_f32(
        /*neg_a=*/false, a, /*neg_b=*/false, b,
        /*c_mod=*/(short)0, c, /*reuse_a=*/false, /*reuse_b=*/false);
}

// ---------------------------------------------------------------- utilities
__global__ void fill_f32(float* __restrict__ p, float v, int n) {
    int i = blockIdx.x * blockDim.x + threadIdx.x;
    if (i < n) p[i] = v;
}

__global__ void degree_kernel(const int* __restrict__ col, float* __restrict__ deg, int e) {
    int i = blockIdx.x * blockDim.x + threadIdx.x;
    if (i < e) atomicAdd(&deg[col[i]], 1.0f);
}

__global__ void dinv_kernel(const float* __restrict__ deg, float* __restrict__ dinv, int n) {
    int i = blockIdx.x * blockDim.x + threadIdx.x;
    if (i < n) {
        float d = deg[i];
        dinv[i] = (d > 0.0f) ? rsqrtf(d) : 0.0f;
    }
}

// ------------------------------------------------- GEMM1: h1 = x @ W1  [N,64]
// block = 128 threads = 4 waves; each wave produces one 16x64 row tile.
// W1 staged in LDS in K-paired layout: sW[(k>>1)*2*HID + n*2 + (k&1)], so each
// B fragment {W[kb][n], W[kb+1][n]} is one aligned ds_load_b64 into an even
// VGPR pair -- directly consumable by v_wmma (no repacking movs).
__global__ void gemm1_wmma(const float* __restrict__ x,
                           const float* __restrict__ W1,
                           float* __restrict__ h1, int nTiles) {
    __shared__ float sW[IN_FEAT * HID];           // 32 KB of the 320 KB/WGP LDS
    for (int i = threadIdx.x; i < IN_FEAT * HID; i += blockDim.x) {
        int k = i / HID, nn = i % HID;
        sW[(k >> 1) * (2 * HID) + nn * 2 + (k & 1)] = W1[i];
    }
    __syncthreads();

    const int wave = threadIdx.x >> 5;
    const int lane = threadIdx.x & 31;
    const int rowTile = blockIdx.x * 4 + wave;
    if (rowTile >= nTiles) return;

    const int rowBase = rowTile * 16;
    const int mrow    = rowBase + (lane & 15);     // A-frag row for this lane
    const int koff    = (lane >> 4) * 2;           // K sub-pair per half-wave
    const int nlane   = lane & 15;                 // B/C column within tile

    const float* xrow = x + (size_t)mrow * IN_FEAT;

    v8f c0 = {}, c1 = {}, c2 = {}, c3 = {};
    for (int kk = 0; kk < IN_FEAT; kk += 4) {
        const int kb = kk + koff;                  // even
        v2f a = *reinterpret_cast<const v2f*>(xrow + kb);   // A[m][kb], A[m][kb+1]
        const float* wp = sW + (kb >> 1) * (2 * HID) + nlane * 2;
        v2f b0 = *reinterpret_cast<const v2f*>(wp + 0);     // cols  0..15
        v2f b1 = *reinterpret_cast<const v2f*>(wp + 32);    // cols 16..31
        v2f b2 = *reinterpret_cast<const v2f*>(wp + 64);    // cols 32..47
        v2f b3 = *reinterpret_cast<const v2f*>(wp + 96);    // cols 48..63
        c0 = wmma_f32_k4(a, b0, c0);
        c1 = wmma_f32_k4(a, b1, c1);
        c2 = wmma_f32_k4(a, b2, c2);
        c3 = wmma_f32_k4(a, b3, c3);
    }

    const int rowHalf = (lane >> 4) * 8;           // C layout: lanes16-31 hold M+8
#pragma unroll
    for (int r = 0; r < 8; ++r) {
        size_t base = (size_t)(rowBase + rowHalf + r) * HID + nlane;
        h1[base +  0] = c0[r];
        h1[base + 16] = c1[r];
        h1[base + 32] = c2[r];
        h1[base + 48] = c3[r];
    }
}

// ---------------------------------- scatter1: agg1[col] += h1[row]*norm  (64f)
// 16 threads per message, float4 per thread.
__global__ void scatter1_kernel(const float* __restrict__ h1,
                                const int* __restrict__ ei,   // [2*E] row|col
                                const float* __restrict__ dinv,
                                float* __restrict__ agg, int e, int n) {
    int tid = blockIdx.x * blockDim.x + threadIdx.x;
    int m  = tid >> 4;
    int fo = (tid & 15) * 4;
    if (m >= e + n) return;
    int src, dst;
    if (m < e) { src = ei[m]; dst = ei[e + m]; }
    else       { src = dst = m - e; }               // self loop
    float w = dinv[src] * dinv[dst];
    float4 v = *reinterpret_cast<const float4*>(h1 + (size_t)src * HID + fo);
    float* p = agg + (size_t)dst * HID + fo;
    atomicAdd(p + 0, v.x * w);
    atomicAdd(p + 1, v.y * w);
    atomicAdd(p + 2, v.z * w);
    atomicAdd(p + 3, v.w * w);
}

__global__ void bias_relu_kernel(float* __restrict__ a, const float* __restrict__ b1, int total) {
    int i = blockIdx.x * blockDim.x + threadIdx.x;
    if (i < total) {
        float v = a[i] + b1[i & (HID - 1)];
        a[i] = v > 0.0f ? v : 0.0f;
    }
}

// ------------------------------------------------- GEMM2: h3 = h2 @ W2 [N,40]
// W2 zero-padded to 64x48 in LDS, same K-paired layout as GEMM1.
__global__ void gemm2_wmma(const float* __restrict__ h2,
                           const float* __restrict__ W2,
                           float* __restrict__ h3, int nTiles) {
    __shared__ float sW[HID * CLS_PAD];           // 12 KB, K-paired
    for (int i = threadIdx.x; i < HID * CLS_PAD; i += blockDim.x) {
        int k = i / CLS_PAD, nn = i % CLS_PAD;
        sW[(k >> 1) * (2 * CLS_PAD) + nn * 2 + (k & 1)] =
            (nn < CLS) ? W2[k * CLS + nn] : 0.0f;
    }
    __syncthreads();

    const int wave = threadIdx.x >> 5;
    const int lane = threadIdx.x & 31;
    const int rowTile = blockIdx.x * 4 + wave;
    if (rowTile >= nTiles) return;

    const int rowBase = rowTile * 16;
    const int mrow    = rowBase + (lane & 15);
    const int koff    = (lane >> 4) * 2;
    const int nlane   = lane & 15;

    const float* hrow = h2 + (size_t)mrow * HID;

    v8f c0 = {}, c1 = {}, c2 = {};
    for (int kk = 0; kk < HID; kk += 4) {
        const int kb = kk + koff;                  // even
        v2f a = *reinterpret_cast<const v2f*>(hrow + kb);
        const float* wp = sW + (kb >> 1) * (2 * CLS_PAD) + nlane * 2;
        v2f b0 = *reinterpret_cast<const v2f*>(wp + 0);     // cols  0..15
        v2f b1 = *reinterpret_cast<const v2f*>(wp + 32);    // cols 16..31
        v2f b2 = *reinterpret_cast<const v2f*>(wp + 64);    // cols 32..47 (pad)
        c0 = wmma_f32_k4(a, b0, c0);
        c1 = wmma_f32_k4(a, b1, c1);
        c2 = wmma_f32_k4(a, b2, c2);
    }

    const int rowHalf = (lane >> 4) * 8;
#pragma unroll
    for (int r = 0; r < 8; ++r) {
        size_t base = (size_t)(rowBase + rowHalf + r) * CLS + nlane;
        h3[base +  0] = c0[r];
        h3[base + 16] = c1[r];
        if (nlane < CLS - 32) h3[base + 32] = c2[r];   // cols 32..39 only
    }
}

// ---------------------------------- scatter2: agg2[col] += h3[row]*norm (40f)
// 10 threads per message, float4 per thread.
__global__ void scatter2_kernel(const float* __restrict__ h3,
                                const int* __restrict__ ei,
                                const float* __restrict__ dinv,
                                float* __restrict__ agg, int e, int n) {
    int tid = blockIdx.x * blockDim.x + threadIdx.x;
    int m  = tid / 10;
    int fo = (tid % 10) * 4;
    if (m >= e + n) return;
    int src, dst;
    if (m < e) { src = ei[m]; dst = ei[e + m]; }
    else       { src = dst = m - e; }
    float w = dinv[src] * dinv[dst];
    float4 v = *reinterpret_cast<const float4*>(h3 + (size_t)src * CLS + fo);
    float* p = agg + (size_t)dst * CLS + fo;
    atomicAdd(p + 0, v.x * w);
    atomicAdd(p + 1, v.y * w);
    atomicAdd(p + 2, v.z * w);
    atomicAdd(p + 3, v.w * w);
}

// -------------------------------------- per-node bias + log_softmax over 40
__global__ void logsoftmax_kernel(const float* __restrict__ agg,
                                  const float* __restrict__ b2,
                                  float* __restrict__ out, int n) {
    int i = blockIdx.x * blockDim.x + threadIdx.x;
    if (i >= n) return;
    float vals[CLS];
    float mx = -3.402823466e+38f;
#pragma unroll
    for (int f = 0; f < CLS; ++f) {
        float v = agg[(size_t)i * CLS + f] + b2[f];
        vals[f] = v;
        mx = v > mx ? v : mx;
    }
    float s = 0.0f;
#pragma unroll
    for (int f = 0; f < CLS; ++f) s += expf(vals[f] - mx);
    float lse = mx + logf(s);
#pragma unroll
    for (int f = 0; f < CLS; ++f) out[(size_t)i * CLS + f] = vals[f] - lse;
}

// ---------------------------------------------------------------------------
extern "C" void kernel_launch(void* const* d_in, const int* in_sizes, int n_in,
                              void* d_out, int out_size, void* d_ws, size_t ws_size,
                              hipStream_t stream) {
    const float* x  = (const float*)d_in[0];   // [N,128]
    const int*   ei = (const int*)d_in[1];     // [2,E] flat: rows then cols
    const float* W1 = (const float*)d_in[2];   // [128,64]
    const float* b1 = (const float*)d_in[3];   // [64]
    const float* W2 = (const float*)d_in[4];   // [64,40]
    const float* b2 = (const float*)d_in[5];   // [40]
    float* out = (float*)d_out;                // [N,40]

    const int N = N_NODES, E = N_EDGES;

    // workspace layout (floats)
    float* deg  = (float*)d_ws;                 // N
    float* dinv = deg  + N;                     // N
    float* h1   = dinv + N;                     // N*64
    float* agg1 = h1   + (size_t)N * HID;       // N*64   (becomes h2 in place)
    float* h3   = agg1 + (size_t)N * HID;       // N*40
    float* agg2 = h3   + (size_t)N * CLS;       // N*40

    const int BT = 256;
    const int nTiles = N / 16;                  // 6250, exact
    const int gemmBlocks = (nTiles + 3) / 4;

    // deg = 1 (self loops) ; agg buffers = 0
    fill_f32<<<(N + BT - 1) / BT, BT, 0, stream>>>(deg, 1.0f, N);
    fill_f32<<<((N * HID) + BT - 1) / BT, BT, 0, stream>>>(agg1, 0.0f, N * HID);
    fill_f32<<<((N * CLS) + BT - 1) / BT, BT, 0, stream>>>(agg2, 0.0f, N * CLS);

    degree_kernel<<<(E + BT - 1) / BT, BT, 0, stream>>>(ei + E, deg, E);
    dinv_kernel<<<(N + BT - 1) / BT, BT, 0, stream>>>(deg, dinv, N);

    gemm1_wmma<<<gemmBlocks, 128, 0, stream>>>(x, W1, h1, nTiles);

    {
        int msgs = (E + N) * 16;                // 16 threads per message
        scatter1_kernel<<<(msgs + BT - 1) / BT, BT, 0, stream>>>(h1, ei, dinv, agg1, E, N);
    }

    bias_relu_kernel<<<((N * HID) + BT - 1) / BT, BT, 0, stream>>>(agg1, b1, N * HID);

    gemm2_wmma<<<gemmBlocks, 128, 0, stream>>>(agg1, W2, h3, nTiles);

    {
        int msgs = (E + N) * 10;                // 10 threads per message
        scatter2_kernel<<<(msgs + BT - 1) / BT, BT, 0, stream>>>(h3, ei, dinv, agg2, E, N);
    }

    logsoftmax_kernel<<<(N + BT - 1) / BT, BT, 0, stream>>>(agg2, b2, out, N);
}